// GATGNN_43379169689776
// MI455X (gfx1250) — compile-verified
//
#include <hip/hip_runtime.h>
#include <hip/hip_bf16.h>

// ---------------------------------------------------------------------------
// 3-layer GAT (128->128->128->128) on 50k nodes / 1.6M edges, fp32 throughout.
// GEMM uses V_WMMA_F32_16X16X4_F32 (true-fp32 matrix op, preserves reference
// numerics). Edge aggregation is wave-per-edge with float4 lanes (wave32*4 = 128).
// Working set (z/h 25.6MB each, edges 12.8MB, e 6.4MB) is L2-resident (192MB).
// ---------------------------------------------------------------------------

typedef __attribute__((ext_vector_type(2))) float v2f;
typedef __attribute__((ext_vector_type(8))) float v8f;

#define DD 128

// ---- float <-> monotonic-uint mapping for atomic float max ----------------
__device__ __forceinline__ unsigned f2o(float f) {
    unsigned u = __float_as_uint(f);
    return (u & 0x80000000u) ? ~u : (u | 0x80000000u);
}
__device__ __forceinline__ float o2f(unsigned u) {
    return (u & 0x80000000u) ? __uint_as_float(u & 0x7fffffffu)
                             : __uint_as_float(~u);
}
#define NEG_INF_ORD 0x007fffffu  // f2o(-inf)

// ---------------------------------------------------------------------------
// K1: Z = act(X) @ W^T + b      X:[N,128]  W:[128,128] (torch Linear layout)
// One block = 8 waves; wave w computes the 16-col tile w of a 16-row strip.
// 32 x V_WMMA_F32_16X16X4_F32 per wave; bias folded into C-init; ReLU is a
// compile-time template parameter so layer 0 pays zero activation overhead.
// ---------------------------------------------------------------------------
template <bool RELU>
__global__ __launch_bounds__(256) void gat_gemm(const float* __restrict__ X,
                                                const float* __restrict__ W,
                                                const float* __restrict__ bias,
                                                float* __restrict__ Z,
                                                int nStrips) {
    const int strip = blockIdx.x;
    if (strip >= nStrips) return;
    const int wave = threadIdx.x >> 5;   // n-tile 0..7
    const int lane = threadIdx.x & 31;
    const int mrow_a = (strip << 4) + (lane & 15);   // row this lane feeds A
    const int khalf  = (lane >> 4) << 1;             // 0 or 2
    const int ncol   = (wave << 4) + (lane & 15);    // output column

    const float* __restrict__ xrow = X + (size_t)mrow_a * DD;
    const float* __restrict__ wrow = W + (size_t)ncol * DD; // B[k][n] = W[n][k]

    // Bias depends only on ncol -> identical across all 8 C registers.
    const float bn = bias[ncol];
    v8f acc = {bn, bn, bn, bn, bn, bn, bn, bn};

#pragma unroll 8
    for (int k0 = 0; k0 < DD; k0 += 4) {
        v2f a, b;
        const v2f av = *(const v2f*)(xrow + k0 + khalf);
        if (RELU) {
            a.x = fmaxf(av.x, 0.f);
            a.y = fmaxf(av.y, 0.f);
        } else {
            a = av;
        }
        b = *(const v2f*)(wrow + k0 + khalf);
        acc = __builtin_amdgcn_wmma_f32_16x16x4_f32(false, a, false, b,
                                                    (short)0, acc, false, false);
    }

    // C/D layout: VGPR r -> (M = r + 8*(lane>=16), N = lane&15 within tile)
    const int mbase = (strip << 4) + ((lane >> 4) << 3);
#pragma unroll
    for (int r = 0; r < 8; ++r)
        Z[(size_t)(mbase + r) * DD + ncol] = acc[r];
}

// ---------------------------------------------------------------------------
// K2: per node: a_src = z.aW[:128], a_dst = z.aW[128:]; init m,s; zero Hout.
// One wave per node; float4 per lane; wave32 shfl_xor reduction.
// ---------------------------------------------------------------------------
__global__ __launch_bounds__(256) void gat_node_prep(const float* __restrict__ Z,
                                                     const float* __restrict__ aW,
                                                     float* __restrict__ a_src,
                                                     float* __restrict__ a_dst,
                                                     unsigned* __restrict__ m_u,
                                                     float* __restrict__ s,
                                                     float* __restrict__ Hout,
                                                     int nNodes) {
    const int node = (blockIdx.x << 3) + (threadIdx.x >> 5);
    const int lane = threadIdx.x & 31;
    if (node >= nNodes) return;

    const float4 zv  = ((const float4*)(Z + (size_t)node * DD))[lane];
    const float4 as4 = ((const float4*)aW)[lane];
    const float4 ad4 = ((const float4*)(aW + DD))[lane];

    float ps = zv.x * as4.x + zv.y * as4.y + zv.z * as4.z + zv.w * as4.w;
    float pd = zv.x * ad4.x + zv.y * ad4.y + zv.z * ad4.z + zv.w * ad4.w;

    ((float4*)(Hout + (size_t)node * DD))[lane] = make_float4(0.f, 0.f, 0.f, 0.f);

#pragma unroll
    for (int off = 16; off > 0; off >>= 1) {
        ps += __shfl_xor(ps, off);
        pd += __shfl_xor(pd, off);
    }
    if (lane == 0) {
        a_src[node] = ps;
        a_dst[node] = pd;
        m_u[node]   = NEG_INF_ORD;
        s[node]     = 0.f;
    }
}

// ---------------------------------------------------------------------------
// K3: e = leaky_relu(a_src[src]+a_dst[dst]+ab); segment max via atomic u32.
// ---------------------------------------------------------------------------
__global__ __launch_bounds__(256) void gat_edge_max(const int* __restrict__ src,
                                                    const int* __restrict__ dst,
                                                    const float* __restrict__ a_src,
                                                    const float* __restrict__ a_dst,
                                                    const float* __restrict__ ab,
                                                    float* __restrict__ e_buf,
                                                    unsigned* __restrict__ m_u,
                                                    int E) {
    const int i = blockIdx.x * blockDim.x + threadIdx.x;
    if (i >= E) return;
    const int d = dst[i];
    float v = a_src[src[i]] + a_dst[d] + ab[0];
    v = (v > 0.f) ? v : 0.01f * v;          // torch leaky_relu default slope
    e_buf[i] = v;
    atomicMax(&m_u[d], f2o(v));
}

// ---------------------------------------------------------------------------
// K4: ex = exp(e - m[dst]); segment sum via atomic f32 add.
// ---------------------------------------------------------------------------
__global__ __launch_bounds__(256) void gat_edge_exp(const int* __restrict__ dst,
                                                    float* __restrict__ e_buf,
                                                    const unsigned* __restrict__ m_u,
                                                    float* __restrict__ s,
                                                    int E) {
    const int i = blockIdx.x * blockDim.x + threadIdx.x;
    if (i >= E) return;
    const int d = dst[i];
    const float ex = expf(e_buf[i] - o2f(m_u[d]));
    e_buf[i] = ex;
    atomicAdd(&s[d], ex);
}

// ---------------------------------------------------------------------------
// K5: h[dst] += (ex/s[dst]) * z[src]   — one wave per edge, float4 per lane.
// ---------------------------------------------------------------------------
__global__ __launch_bounds__(256) void gat_edge_aggr(const int* __restrict__ src,
                                                     const int* __restrict__ dst,
                                                     const float* __restrict__ e_buf,
                                                     const float* __restrict__ s,
                                                     const float* __restrict__ Z,
                                                     float* __restrict__ Hout,
                                                     int E) {
    const int edge = (blockIdx.x << 3) + (threadIdx.x >> 5);
    const int lane = threadIdx.x & 31;
    if (edge >= E) return;
    const int sn = src[edge], dn = dst[edge];
    const float alpha = e_buf[edge] / s[dn];
    const float4 zv = ((const float4*)(Z + (size_t)sn * DD))[lane];
    float* hp = Hout + (size_t)dn * DD + (lane << 2);
    atomicAdd(hp + 0, alpha * zv.x);
    atomicAdd(hp + 1, alpha * zv.y);
    atomicAdd(hp + 2, alpha * zv.z);
    atomicAdd(hp + 3, alpha * zv.w);
}

// ---------------------------------------------------------------------------
// Host-side orchestration
// ---------------------------------------------------------------------------
static inline size_t alignup(size_t v) { return (v + 255) & ~(size_t)255; }

extern "C" void kernel_launch(void* const* d_in, const int* in_sizes, int n_in,
                              void* d_out, int out_size, void* d_ws, size_t ws_size,
                              hipStream_t stream) {
    const float* x   = (const float*)d_in[0];
    const int*   src = (const int*)d_in[1];
    const int*   dst = (const int*)d_in[2];
    // d_in[3] = t (unused)
    const int N = in_sizes[0] / DD;     // 50000 (multiple of 16)
    const int E = in_sizes[1];          // 1600000

    // workspace carve-up
    char* ws = (char*)d_ws;
    size_t off = 0;
    float* Z  = (float*)(ws + off); off = alignup(off + (size_t)N * DD * 4);
    float* H1 = (float*)(ws + off); off = alignup(off + (size_t)N * DD * 4);
    float* H2 = (float*)(ws + off); off = alignup(off + (size_t)N * DD * 4);
    float* As = (float*)(ws + off); off = alignup(off + (size_t)N * 4);
    float* Ad = (float*)(ws + off); off = alignup(off + (size_t)N * 4);
    unsigned* Mu = (unsigned*)(ws + off); off = alignup(off + (size_t)N * 4);
    float* S  = (float*)(ws + off); off = alignup(off + (size_t)N * 4);
    float* Eb = (float*)(ws + off); off = alignup(off + (size_t)E * 4);
    (void)ws_size;

    const int nStrips   = N / 16;                 // 3125
    const int nodeBlks  = (N + 7) / 8;            // 8 waves/block
    const int edgeBlks  = (E + 255) / 256;
    const int edgeWaveB = (E + 7) / 8;            // 1 wave/edge

    const float* layer_in[3]  = { x,  H1, H2 };
    float*       layer_out[3] = { H1, H2, (float*)d_out };

    for (int L = 0; L < 3; ++L) {
        const float* W  = (const float*)d_in[4 + 4 * L];
        const float* b  = (const float*)d_in[5 + 4 * L];
        const float* aW = (const float*)d_in[6 + 4 * L];
        const float* ab = (const float*)d_in[7 + 4 * L];
        float* Hout = layer_out[L];

        if (L == 0)
            gat_gemm<false><<<nStrips, 256, 0, stream>>>(layer_in[L], W, b, Z, nStrips);
        else
            gat_gemm<true><<<nStrips, 256, 0, stream>>>(layer_in[L], W, b, Z, nStrips);
        gat_node_prep<<<nodeBlks, 256, 0, stream>>>(Z, aW, As, Ad, Mu, S, Hout, N);
        gat_edge_max<<<edgeBlks, 256, 0, stream>>>(src, dst, As, Ad, ab, Eb, Mu, E);
        gat_edge_exp<<<edgeBlks, 256, 0, stream>>>(dst, Eb, Mu, S, E);
        gat_edge_aggr<<<edgeWaveB, 256, 0, stream>>>(src, dst, Eb, S, Z, Hout, E);
    }
}